// LinearInterpolate_1219770712097
// MI455X (gfx1250) — compile-verified
//
#include <hip/hip_runtime.h>
#include <stdint.h>

// Problem constants (match the reference: B, T, D = 32, 2048, 128; fp32).
#define B_  32
#define T_  2048
#define D_  128
#define CT_ 16              // timesteps per chunk
#define NC_ (T_ / CT_)      // 128 chunks

// ---------------------------------------------------------------------------
// Pass 1: per-(batch, chunk, feature) summary of the mask:
//   lastObs  = last observed t in chunk  (-1 if none)
//   firstObs = first observed t in chunk (T_ if none)
// ---------------------------------------------------------------------------
__global__ void lint_chunk_summary(const int* __restrict__ mask,
                                   int* __restrict__ lastObs,
                                   int* __restrict__ firstObs) {
  const int bc = blockIdx.x;
  const int b  = bc / NC_;
  const int c  = bc % NC_;
  const int d  = threadIdx.x;           // lane = feature
  const int t0 = c * CT_;
  int first = -1, last = -1;
  const int* mp = mask + (size_t)(b * T_ + t0) * D_ + d;
#pragma unroll
  for (int r = 0; r < CT_; ++r) {
    if (mp[(size_t)r * D_] != 0) {
      if (first < 0) first = t0 + r;
      last = t0 + r;
    }
  }
  const int idx = (b * NC_ + c) * D_ + d;
  lastObs[idx]  = last;
  firstObs[idx] = (first < 0) ? T_ : first;
}

// ---------------------------------------------------------------------------
// Pass 2: serial carry resolution across the 128 chunks of each column,
// IN PLACE:
//   lastObs[b,c,d]  -> index of last observation in chunks  < c  (-1 if none)
//   firstObs[b,c,d] -> index of first observation in chunks > c  (T_ if none)
// ---------------------------------------------------------------------------
__global__ void lint_resolve(int* __restrict__ lastObs,
                             int* __restrict__ firstObs) {
  const int b = blockIdx.x;
  const int d = threadIdx.x;
  int fcarry = -1;
  for (int c = 0; c < NC_; ++c) {
    const int idx = (b * NC_ + c) * D_ + d;
    const int tmp = lastObs[idx];
    lastObs[idx]  = fcarry;
    if (tmp >= 0) fcarry = tmp;
  }
  int bcarry = T_;
  for (int c = NC_ - 1; c >= 0; --c) {
    const int idx = (b * NC_ + c) * D_ + d;
    const int tmp = firstObs[idx];
    firstObs[idx] = bcarry;
    if (tmp < T_) bcarry = tmp;
  }
}

// ---------------------------------------------------------------------------
// Pass 3: per-(batch, chunk) block. Stage the chunk's values/times/mask into
// LDS via gfx1250 async global->LDS copies (ASYNCcnt), run the backward fill
// into registers, then the forward fill + interpolation. Output is written
// with non-temporal stores so the 33.5 MB write-once result does not evict
// the L2-resident values/times arrays needed by the carry-seed gathers.
// LDS: 3 * CT_*D_ * 4B = 24 KB per block.
// ---------------------------------------------------------------------------
__global__ void lint_interp(const float* __restrict__ values,
                            const float* __restrict__ times,
                            const int*   __restrict__ mask,
                            const int*   __restrict__ fwdC,   // resolved lastObs
                            const int*   __restrict__ bwdC,   // resolved firstObs
                            float* __restrict__ out) {
  __shared__ __align__(16) float s_val[CT_ * D_];
  __shared__ __align__(16) float s_tim[CT_ * D_];
  __shared__ __align__(16) int   s_msk[CT_ * D_];

  const int bc = blockIdx.x;
  const int b  = bc / NC_;
  const int c  = bc % NC_;
  const int d  = threadIdx.x;
  const int t0 = c * CT_;
  const size_t chunkBase = (size_t)(b * T_ + t0) * D_;   // contiguous 8 KB slab

  // ---- async stage: 3 arrays x 4 x b128 per thread (12 async ops / lane) ----
  {
    const uint64_t gv = (uint64_t)(uintptr_t)(values + chunkBase);
    const uint64_t gt = (uint64_t)(uintptr_t)(times  + chunkBase);
    const uint64_t gm = (uint64_t)(uintptr_t)(mask   + chunkBase);
    const uint32_t lv = (uint32_t)(uintptr_t)&s_val[0];
    const uint32_t lt = (uint32_t)(uintptr_t)&s_tim[0];
    const uint32_t lm = (uint32_t)(uintptr_t)&s_msk[0];
#pragma unroll
    for (int k = 0; k < (CT_ * D_) / (4 * 128); ++k) {     // 4 iters
      const uint32_t off = (uint32_t)(k * 128 + threadIdx.x) * 16u;
      uint32_t l; uint64_t g;
      l = lv + off; g = gv + off;
      asm volatile("global_load_async_to_lds_b128 %0, %1, off"
                   :: "v"(l), "v"(g) : "memory");
      l = lt + off; g = gt + off;
      asm volatile("global_load_async_to_lds_b128 %0, %1, off"
                   :: "v"(l), "v"(g) : "memory");
      l = lm + off; g = gm + off;
      asm volatile("global_load_async_to_lds_b128 %0, %1, off"
                   :: "v"(l), "v"(g) : "memory");
    }
    asm volatile("s_wait_asynccnt 0x0" ::: "memory");
  }
  __syncthreads();

  // ---- seed carries from resolved cross-chunk indices (L2-resident gathers) --
  const int cidx = (b * NC_ + c) * D_ + d;
  const int fc  = __builtin_nontemporal_load(fwdC + cidx);  // read exactly once
  const int bcv = __builtin_nontemporal_load(bwdC + cidx);  // read exactly once

  float xl, tl;
  if (fc >= 0) {
    const size_t gi = (size_t)(b * T_ + fc) * D_ + d;
    xl = values[gi]; tl = times[gi];
  } else {
    xl = 0.0f; tl = times[(size_t)(b * T_) * D_ + d];          // times[b,0,d]
  }
  float xn, tn;
  if (bcv < T_) {
    const size_t gi = (size_t)(b * T_ + bcv) * D_ + d;
    xn = values[gi]; tn = times[gi];
  } else {
    xn = 0.0f; tn = times[(size_t)(b * T_ + (T_ - 1)) * D_ + d]; // axis max
  }

  // ---- backward fill within chunk (inclusive), kept in registers ----
  float xn_r[CT_], tn_r[CT_];
#pragma unroll
  for (int r = CT_ - 1; r >= 0; --r) {
    const int li = r * D_ + d;
    if (s_msk[li] != 0) { xn = s_val[li]; tn = s_tim[li]; }
    xn_r[r] = xn; tn_r[r] = tn;
  }

  // ---- forward fill + interpolation; streaming NT stores for the output ----
#pragma unroll
  for (int r = 0; r < CT_; ++r) {
    const int li = r * D_ + d;
    const float v  = s_val[li];
    const float tt = s_tim[li];
    const bool  m  = (s_msk[li] != 0);
    if (m) { xl = v; tl = tt; }
    const float tnn = tn_r[r];
    const float xnn = xn_r[r];
    const float denom = tnn - tl;
    const bool  safe  = (denom != 0.0f);
    const float num   = xl * (tnn - tt) + xnn * (tt - tl);
    float xi = num / (safe ? denom : 1.0f);
    // isfinite(): NaN/Inf comparisons are false, matching the reference guard
    xi = (safe && (__builtin_fabsf(xi) <= 3.402823466e38f)) ? xi : 0.0f;
    __builtin_nontemporal_store(m ? v : xi, out + chunkBase + (size_t)r * D_ + d);
  }
}

// ---------------------------------------------------------------------------
// kernel_launch: inputs in reference setup order: values(f32), times(f32),
// mask(int). Output: f32, B*T*D. Workspace use: 2 * B*NC*D int32 = 4 MB.
// ---------------------------------------------------------------------------
extern "C" void kernel_launch(void* const* d_in, const int* in_sizes, int n_in,
                              void* d_out, int out_size, void* d_ws, size_t ws_size,
                              hipStream_t stream) {
  (void)in_sizes; (void)n_in; (void)out_size; (void)ws_size;
  const float* values = (const float*)d_in[0];
  const float* times  = (const float*)d_in[1];
  const int*   mask   = (const int*)d_in[2];
  float* out = (float*)d_out;

  int* lastObs  = (int*)d_ws;                   // B_*NC_*D_ ints (2 MB)
  int* firstObs = lastObs + (B_ * NC_ * D_);    // B_*NC_*D_ ints (2 MB)

  lint_chunk_summary<<<B_ * NC_, D_, 0, stream>>>(mask, lastObs, firstObs);
  lint_resolve      <<<B_,        D_, 0, stream>>>(lastObs, firstObs);
  lint_interp       <<<B_ * NC_, D_, 0, stream>>>(values, times, mask,
                                                  lastObs, firstObs, out);
}